// InstanceGroupItN_29222957482782
// MI455X (gfx1250) — compile-verified
//
#include <hip/hip_runtime.h>
#include <hip/hip_bf16.h>

typedef __attribute__((ext_vector_type(2))) float v2f;
typedef __attribute__((ext_vector_type(8))) float v8f;

#define B_BATCH 32
#define GROUPS 32
#define MLEN 65536            // (C/G)*L = 16*4096
#define BATCH_STRIDE 2097152  // 512*4096

// Async copy: one 16B chunk, global -> LDS, tracked by ASYNCcnt.
__device__ __forceinline__ void async_ld16(const float* g, void* l) {
  unsigned lds = (unsigned)(uintptr_t)l;  // low 32 bits of flat shared ptr = LDS offset
  asm volatile("global_load_async_to_lds_b128 %0, %1, off" :: "v"(lds), "v"(g) : "memory");
}
__device__ __forceinline__ void wait_async_le4() {
  asm volatile("s_wait_asynccnt 0x4" ::: "memory");
}
__device__ __forceinline__ void wait_async_0() {
  asm volatile("s_wait_asynccnt 0x0" ::: "memory");
}

// ---------------- Kernel 1: partial gram + row sums (WMMA f32 16x16x4) ------
// 16 WGs per batch, each covers 4096 columns of the 32 x 65536 group matrix.
// Double-buffered async staging of 32x128 tiles.
#define K1_TK 128
#define K1_STRIDE 132   // 132 % 64 == 4 -> conflict-free b64 A/B reads

__global__ __launch_bounds__(256) void ign_k1_stats(
    const float* __restrict__ x, float* __restrict__ pgram, float* __restrict__ psum) {
  const int b  = blockIdx.x >> 4;
  const int wg = blockIdx.x & 15;
  const int t  = threadIdx.x;
  const int lane = t & 31;
  const int wv   = t >> 5;

  __shared__ union {
    float tile[2][32][K1_STRIDE];
    struct { float pgm[8][1024]; float rs[32][16]; } red;
  } s;

  const float* xb = x + (size_t)b * BATCH_STRIDE + (size_t)wg * 4096;

  auto stage = [&](int tilei, int bi) {
#pragma unroll
    for (int i = 0; i < 4; ++i) {
      const int idx = i * 256 + t;
      const int row = idx >> 5;          // 0..31, 32 lanes per row (512B coalesced)
      const int col = (idx & 31) * 4;    // 0..124
      async_ld16(xb + (size_t)row * MLEN + tilei * K1_TK + col, &s.tile[bi][row][col]);
    }
  };

  v8f c00 = {}, c01 = {}, c10 = {}, c11 = {};
  float rs_lo = 0.0f, rs_hi = 0.0f;

  const int arow = lane & 15;          // M (and N) index
  const int koff = 2 * (lane >> 4);    // upper half-wave holds K+2,K+3

  stage(0, 0);
  for (int tl = 0; tl < 32; ++tl) {
    const int bi = tl & 1;
    if (tl + 1 < 32) { stage(tl + 1, bi ^ 1); wait_async_le4(); }
    else             { wait_async_0(); }
    __syncthreads();

    const int cbase = wv * 16;  // wave-private 16-column slice -> 4 K-steps
#pragma unroll
    for (int kk = 0; kk < 4; ++kk) {
      const int k0 = cbase + kk * 4 + koff;
      v2f rlo = *(const v2f*)&s.tile[bi][arow][k0];       // rows 0..15
      v2f rhi = *(const v2f*)&s.tile[bi][16 + arow][k0];  // rows 16..31
      rs_lo += rlo.x + rlo.y;   // each (row,k) touched exactly once per wave
      rs_hi += rhi.x + rhi.y;
      // same registers serve as A (16x4) and B (4x16 of X^T)
      c00 = __builtin_amdgcn_wmma_f32_16x16x4_f32(false, rlo, false, rlo, (short)0, c00, false, false);
      c01 = __builtin_amdgcn_wmma_f32_16x16x4_f32(false, rlo, false, rhi, (short)0, c01, false, false);
      c10 = __builtin_amdgcn_wmma_f32_16x16x4_f32(false, rhi, false, rlo, (short)0, c10, false, false);
      c11 = __builtin_amdgcn_wmma_f32_16x16x4_f32(false, rhi, false, rhi, (short)0, c11, false, false);
    }
    __syncthreads();
  }

  // dump per-wave gram partials + row-sum partials into LDS, reduce across waves
  const int gl = lane & 15;
  const int hi = lane >> 4;
#pragma unroll
  for (int v = 0; v < 8; ++v) {
    const int g = v + 8 * hi;
    s.red.pgm[wv][g * 32 + gl]              = c00[v];
    s.red.pgm[wv][g * 32 + 16 + gl]         = c01[v];
    s.red.pgm[wv][(16 + g) * 32 + gl]       = c10[v];
    s.red.pgm[wv][(16 + g) * 32 + 16 + gl]  = c11[v];
  }
  s.red.rs[arow][wv * 2 + hi]      = rs_lo;
  s.red.rs[16 + arow][wv * 2 + hi] = rs_hi;
  __syncthreads();

  float* gout = pgram + (size_t)(b * 16 + wg) * 1024;
  for (int e = t; e < 1024; e += 256) {
    float sum = 0.0f;
#pragma unroll
    for (int w = 0; w < 8; ++w) sum += s.red.pgm[w][e];
    gout[e] = sum;
  }
  if (t < 32) {
    float sum = 0.0f;
#pragma unroll
    for (int k = 0; k < 16; ++k) sum += s.red.rs[t][k];
    psum[(b * 16 + wg) * 32 + t] = sum;
  }
}

// ---------------- Kernel 2: reduce + Newton-Schulz on 32x32 -----------------
__device__ __forceinline__ void mm32(float (*D)[33], const float (*A)[33],
                                     const float (*Bm)[33], int t) {
  __syncthreads();
  float r[4];
#pragma unroll
  for (int q = 0; q < 4; ++q) {
    const int e = t + q * 256;
    const int i = e >> 5, j = e & 31;
    float sacc = 0.0f;
#pragma unroll
    for (int k = 0; k < 32; ++k) sacc += A[i][k] * Bm[k][j];
    r[q] = sacc;
  }
  __syncthreads();
#pragma unroll
  for (int q = 0; q < 4; ++q) {
    const int e = t + q * 256;
    D[e >> 5][e & 31] = r[q];
  }
  __syncthreads();
}

__global__ __launch_bounds__(256) void ign_k2_newton(
    const float* __restrict__ pgram, const float* __restrict__ psum,
    float* __restrict__ wm, float* __restrict__ off) {
  const int b = blockIdx.x;
  const int t = threadIdx.x;
  __shared__ float sig[32][33], P[32][33], T[32][33], Q[32][33];
  __shared__ float mean[32];
  __shared__ float tinv, sqti;
  const float invm = 1.0f / (float)MLEN;

  for (int e = t; e < 1024; e += 256) {
    float sacc = 0.0f;
    for (int w = 0; w < 16; ++w) sacc += pgram[(size_t)(b * 16 + w) * 1024 + e];
    sig[e >> 5][e & 31] = sacc;
  }
  if (t < 32) {
    float sacc = 0.0f;
    for (int w = 0; w < 16; ++w) sacc += psum[(b * 16 + w) * 32 + t];
    mean[t] = sacc * invm;
  }
  __syncthreads();
  if (t == 0) {
    float tr = 0.0f;
    for (int g = 0; g < 32; ++g) tr += sig[g][g] * invm - mean[g] * mean[g];
    tinv = 1.0f / tr;
    sqti = sqrtf(tinv);
  }
  __syncthreads();
  for (int e = t; e < 1024; e += 256) {
    const int i = e >> 5, j = e & 31;
    sig[i][j] = (sig[i][j] * invm - mean[i] * mean[j]) * tinv;  // sigma_n
    P[i][j] = (i == j) ? 1.0f : 0.0f;
  }
  __syncthreads();

  for (int it = 0; it < 5; ++it) {
    mm32(T, P, P, t);    // T = P^2
    mm32(Q, T, P, t);    // Q = P^3
    mm32(T, Q, sig, t);  // T = P^3 * sigma_n
    for (int e = t; e < 1024; e += 256) {
      const int i = e >> 5, j = e & 31;
      P[i][j] = 1.5f * P[i][j] - 0.5f * T[i][j];
    }
    __syncthreads();
  }

  for (int e = t; e < 1024; e += 256)
    wm[(size_t)b * 1024 + e] = P[e >> 5][e & 31] * sqti;
  if (t < 32) {
    float sacc = 0.0f;
    for (int h = 0; h < 32; ++h) sacc += P[t][h] * mean[h];
    off[b * 32 + t] = sacc * sqti;  // wm @ mean
  }
}

// ---------------- Kernel 3: apply whitening (WMMA) + scale/bias -------------
// 256 WGs per batch; each WG handles 2 blocks of 128 columns, double-buffered.
#define K3_TK 128
#define K3_STRIDE 136  // 2*136 % 64 == 16 -> upper-halfwave B reads disjoint banks

__global__ __launch_bounds__(256) void ign_k3_apply(
    const float* __restrict__ x, const float* __restrict__ wm,
    const float* __restrict__ off, const float* __restrict__ wei,
    const float* __restrict__ bia, float* __restrict__ out) {
  const int b   = blockIdx.x >> 8;   // 256 WGs per batch
  const int wgc = blockIdx.x & 255;  // -> columns [wgc*256, wgc*256+256)
  const int t   = threadIdx.x;
  const int lane = t & 31;
  const int wv   = t >> 5;

  __shared__ float xt[2][32][K3_STRIDE];
  __shared__ float wl[512], bl[512], ofl[32];

  const float* xb = x + (size_t)b * BATCH_STRIDE;
  auto stage = [&](int blk, int bi) {
    const int colbase = (wgc * 2 + blk) * K3_TK;
#pragma unroll
    for (int i = 0; i < 4; ++i) {
      const int idx = i * 256 + t;
      const int row = idx >> 5;
      const int col = (idx & 31) * 4;
      async_ld16(xb + (size_t)row * MLEN + colbase + col, &xt[bi][row][col]);
    }
  };

  stage(0, 0);

  wl[t] = wei[t];       wl[256 + t] = wei[256 + t];
  bl[t] = bia[t];       bl[256 + t] = bia[256 + t];
  if (t < 32) ofl[t] = off[b * 32 + t];

  // preload wm A-tiles into registers (16 float2 per lane)
  const float* wmb = wm + (size_t)b * 1024;
  const int arow = lane & 15;
  const int hk   = 2 * (lane >> 4);
  v2f a[2][8];
#pragma unroll
  for (int kk = 0; kk < 8; ++kk) {
    a[0][kk] = *(const v2f*)&wmb[arow * 32 + kk * 4 + hk];
    a[1][kk] = *(const v2f*)&wmb[(16 + arow) * 32 + kk * 4 + hk];
  }

  float* ob = out + (size_t)b * BATCH_STRIDE;
  const int cl = lane & 15;
  const int hi = lane >> 4;

  for (int blk = 0; blk < 2; ++blk) {
    const int bi = blk;
    if (blk == 0) { stage(1, 1); wait_async_le4(); }
    else          { wait_async_0(); }
    __syncthreads();

    const int tcol = wv * 16;  // one 16-column tile per wave
    v8f c0 = {}, c1 = {};
#pragma unroll
    for (int kk = 0; kk < 8; ++kk) {
      const int kr = kk * 4 + hk;  // B-matrix K rows for this half-wave
      v2f bv;
      bv.x = xt[bi][kr][tcol + cl];
      bv.y = xt[bi][kr + 1][tcol + cl];
      c0 = __builtin_amdgcn_wmma_f32_16x16x4_f32(false, a[0][kk], false, bv, (short)0, c0, false, false);
      c1 = __builtin_amdgcn_wmma_f32_16x16x4_f32(false, a[1][kk], false, bv, (short)0, c1, false, false);
    }
    const int j  = (wgc * 2 + blk) * K3_TK + tcol + cl;  // column in [0, 65536)
    const int jl = j >> 12;                              // j / 4096
#pragma unroll
    for (int v = 0; v < 8; ++v) {
      const int g0 = v + 8 * hi;
      const int g1 = 16 + g0;
      const int ch0 = g0 * 16 + jl;
      const int ch1 = g1 * 16 + jl;
      ob[(size_t)g0 * MLEN + j] = (c0[v] - ofl[g0]) * wl[ch0] + bl[ch0];
      ob[(size_t)g1 * MLEN + j] = (c1[v] - ofl[g1]) * wl[ch1] + bl[ch1];
    }
    __syncthreads();
  }
}

// ---------------------------------------------------------------------------
extern "C" void kernel_launch(void* const* d_in, const int* in_sizes, int n_in,
                              void* d_out, int out_size, void* d_ws, size_t ws_size,
                              hipStream_t stream) {
  const float* x      = (const float*)d_in[0];
  const float* weight = (const float*)d_in[1];
  const float* bias   = (const float*)d_in[2];
  float* out = (float*)d_out;

  float* ws    = (float*)d_ws;
  float* pgram = ws;                         // 32*16*1024 floats
  float* psum  = pgram + 32 * 16 * 1024;     // 32*16*32
  float* wm    = psum + 32 * 16 * 32;        // 32*1024
  float* off   = wm + 32 * 1024;             // 32*32

  ign_k1_stats<<<dim3(32 * 16), dim3(256), 0, stream>>>(x, pgram, psum);
  ign_k2_newton<<<dim3(32), dim3(256), 0, stream>>>(pgram, psum, wm, off);
  ign_k3_apply<<<dim3(32 * 256), dim3(256), 0, stream>>>(x, wm, off, weight, bias, out);
}